// DecoderCell_14791867367481
// MI455X (gfx1250) — compile-verified
//
#include <hip/hip_runtime.h>
#include <math.h>

typedef float v2f __attribute__((ext_vector_type(2)));
typedef float v8f __attribute__((ext_vector_type(8)));

#define Bn 64
#define Tn 2048
#define En 256
#define An 128
#define Rn 1024
#define Dn 1024
#define CTX_SPLIT 8
#define GEMM_SPLIT 4

__device__ __forceinline__ float sigmoidf(float x) {
    return 1.0f / (1.0f + __expf(-x));
}

__device__ __forceinline__ v2f ldv2(const float* p) { return *(const v2f*)p; }

#define WMMA4(A0, A1, A2, A3, Bf)                                                                     \
    acc0 = __builtin_amdgcn_wmma_f32_16x16x4_f32(false, A0, false, Bf, (short)0, acc0, false, false); \
    acc1 = __builtin_amdgcn_wmma_f32_16x16x4_f32(false, A1, false, Bf, (short)0, acc1, false, false); \
    acc2 = __builtin_amdgcn_wmma_f32_16x16x4_f32(false, A2, false, Bf, (short)0, acc2, false, false); \
    acc3 = __builtin_amdgcn_wmma_f32_16x16x4_f32(false, A3, false, Bf, (short)0, acc3, false, false)

// ---------------------------------------------------------------------------
// 2-deep software-pipelined K-range within one weight segment.
// KTOT is compile-time so the 4 A-row offsets are instruction immediates and
// only two pointers advance per step. Two load groups stay in flight, so the
// WMMAs wait only on the older group instead of draining loadcnt to 0.
// ---------------------------------------------------------------------------
template<int KTOT>
__device__ __forceinline__ void gemm_seg(const float* __restrict__ wp,
                                         const float* __restrict__ ap,
                                         int nsteps,
                                         v8f& acc0, v8f& acc1, v8f& acc2, v8f& acc3)
{
    constexpr size_t XR = (size_t)16 * KTOT;   // floats between M-tile rows
    if (nsteps <= 0) return;

    v2f b0  = ldv2(wp);
    v2f a00 = ldv2(ap),          a01 = ldv2(ap + XR);
    v2f a02 = ldv2(ap + 2 * XR), a03 = ldv2(ap + 3 * XR);
    if (nsteps == 1) { WMMA4(a00, a01, a02, a03, b0); return; }

    wp += 4; ap += 4;
    v2f b1  = ldv2(wp);
    v2f a10 = ldv2(ap),          a11 = ldv2(ap + XR);
    v2f a12 = ldv2(ap + 2 * XR), a13 = ldv2(ap + 3 * XR);

    for (int i = 2; i < nsteps; ++i) {
        wp += 4; ap += 4;
        v2f b2  = ldv2(wp);
        v2f a20 = ldv2(ap),          a21 = ldv2(ap + XR);
        v2f a22 = ldv2(ap + 2 * XR), a23 = ldv2(ap + 3 * XR);
        __builtin_prefetch(wp + 256, 0, 1);     // global_prefetch_b8, ~1KB ahead
        WMMA4(a00, a01, a02, a03, b0);
        b0 = b1; a00 = a10; a01 = a11; a02 = a12; a03 = a13;
        b1 = b2; a10 = a20; a11 = a21; a12 = a22; a13 = a23;
    }
    WMMA4(a00, a01, a02, a03, b0);
    WMMA4(a10, a11, a12, a13, b1);
}

// ---------------------------------------------------------------------------
// fp32 WMMA GEMM with K-splitting (all shapes compile-time):
//   out_part[split][64, NN] = X[64, kbeg:kend] * [W0|W1]^T  (+bias on split 0)
// grid = (NN/16, NSPLIT); one wave per block; wave does all 4 M-tiles.
// ---------------------------------------------------------------------------
template<int KTOT, int K0, int NN, int NSPLIT, bool HASW1>
__global__ void __launch_bounds__(32)
gemm_wmma_f32(const float* __restrict__ X,
              const float* __restrict__ W0, const float* __restrict__ W1,
              const float* __restrict__ bias0, const float* __restrict__ bias1,
              float* __restrict__ outp)
{
    const int lane  = threadIdx.x;
    const int l15   = lane & 15;
    const int khalf = lane >> 4;
    const int nbase = blockIdx.x * 16;
    const int nrow  = nbase + l15;

    constexpr int KS = KTOT / NSPLIT;          // multiple of 4 for all call sites
    const int split = blockIdx.y;
    const int kbeg  = split * KS;
    const int kend  = kbeg + KS;
    float* out = outp + (size_t)split * Bn * NN;

    v8f acc0 = {}, acc1 = {}, acc2 = {}, acc3 = {};

    const float* xrow = X + (size_t)l15 * KTOT;

    // segment 0 (W0, X columns [0, K0))
    {
        const int s0b = kbeg < K0 ? kbeg : K0;
        const int s0e = kend < K0 ? kend : K0;
        const float* wp = W0 + (size_t)nrow * K0 + s0b + 2 * khalf;
        const float* ap = xrow + s0b + 2 * khalf;
        gemm_seg<KTOT>(wp, ap, (s0e - s0b) >> 2, acc0, acc1, acc2, acc3);
    }
    // segment 1 (W1, X columns [K0, KTOT))
    if (HASW1) {
        constexpr int K1 = KTOT - K0;
        int s1b = kbeg - K0; if (s1b < 0) s1b = 0;
        int s1e = kend - K0; if (s1e < 0) s1e = 0;
        const float* wp = W1 + (size_t)nrow * K1 + s1b + 2 * khalf;
        const float* ap = xrow + K0 + s1b + 2 * khalf;
        gemm_seg<KTOT>(wp, ap, (s1e - s1b) >> 2, acc0, acc1, acc2, acc3);
    }

    float bn = 0.0f;
    if (split == 0) {
        if (bias0) bn += bias0[nbase + l15];
        if (bias1) bn += bias1[nbase + l15];
    }

    const int col = nbase + l15;
    float* orow = out + (size_t)(8 * khalf) * NN + col;
#pragma unroll
    for (int v = 0; v < 8; ++v) {
        orow[(size_t)( 0 + v) * NN] = acc0[v] + bn;
        orow[(size_t)(16 + v) * NN] = acc1[v] + bn;
        orow[(size_t)(32 + v) * NN] = acc2[v] + bn;
        orow[(size_t)(48 + v) * NN] = acc3[v] + bn;
    }
}

// ---------------------------------------------------------------------------
// dst[i] = sum_s src[s*len + i]
// ---------------------------------------------------------------------------
__global__ void __launch_bounds__(256)
reduce_partials(float* __restrict__ dst, const float* __restrict__ src,
                int nsplit, int len)
{
    int idx = blockIdx.x * blockDim.x + threadIdx.x;
    if (idx >= len) return;
    float s = 0.0f;
    for (int p = 0; p < nsplit; ++p) s += src[(size_t)p * len + idx];
    dst[idx] = s;
}

// ---------------------------------------------------------------------------
// LSTM pointwise over K-split partial gates (i,f,g,o layout).
// ---------------------------------------------------------------------------
__global__ void __launch_bounds__(256)
lstm_pointwise(const float* __restrict__ gates_part, const float* __restrict__ c_in,
               float* __restrict__ h_out, int dim, int nsplit)
{
    int idx = blockIdx.x * blockDim.x + threadIdx.x;
    if (idx >= Bn * dim) return;
    int b = idx / dim, j = idx % dim;
    const size_t stride = (size_t)Bn * 4 * dim;
    const size_t base   = (size_t)b * 4 * dim;
    float gi = 0.f, gf = 0.f, gg = 0.f, go = 0.f;
    for (int s = 0; s < nsplit; ++s) {
        const float* g = gates_part + s * stride + base;
        gi += g[j];
        gf += g[dim + j];
        gg += g[2 * dim + j];
        go += g[3 * dim + j];
    }
    gi = sigmoidf(gi); gf = sigmoidf(gf); gg = tanhf(gg); go = sigmoidf(go);
    float c2 = gf * c_in[(size_t)b * dim + j] + gi * gg;
    h_out[(size_t)b * dim + j] = go * tanhf(c2);
}

// ---------------------------------------------------------------------------
// pack3: dst[64, na+nb+nc] = [A | B | C] (C may be null with nc==0)
// ---------------------------------------------------------------------------
__global__ void __launch_bounds__(256)
pack3(float* __restrict__ dst,
      const float* __restrict__ A, int na,
      const float* __restrict__ Bp, int nb,
      const float* __restrict__ Cp, int nc)
{
    int tot = na + nb + nc;
    int idx = blockIdx.x * blockDim.x + threadIdx.x;
    if (idx >= Bn * tot) return;
    int b = idx / tot, j = idx % tot;
    float v;
    if (j < na)           v = A[(size_t)b * na + j];
    else if (j < na + nb) v = Bp[(size_t)b * nb + (j - na)];
    else                  v = Cp[(size_t)b * nc + (j - na - nb)];
    dst[idx] = v;
}

// ---------------------------------------------------------------------------
// Fused location conv + location FC (WMMA 16x16x4 over K=F=32) + energy.
// One wave per (b, 16-t tile); processed_memory streamed exactly once.
// ---------------------------------------------------------------------------
__global__ void __launch_bounds__(32)
attn_energy(const float* __restrict__ att_w, const float* __restrict__ att_w_cum,
            const float* __restrict__ W_loc, const float* __restrict__ W_lfc,
            const float* __restrict__ pq, const float* __restrict__ pm,
            const float* __restrict__ Wv, const int* __restrict__ mask,
            float* __restrict__ energies)
{
    __shared__ float locS[16 * 32];   // [t_local][f]
    const int b    = blockIdx.y;
    const int t0   = blockIdx.x * 16;
    const int lane = threadIdx.x;

    // ---- conv: lane = filter f ----
    {
        const float* w0  = W_loc + lane * 62;       // [f][0][k]
        const float* w1  = w0 + 31;                 // [f][1][k]
        const float* aw  = att_w     + (size_t)b * Tn;
        const float* awc = att_w_cum + (size_t)b * Tn;
        for (int tl = 0; tl < 16; ++tl) {
            const int t = t0 + tl;
            float s = 0.0f;
#pragma unroll
            for (int k = 0; k < 31; ++k) {
                const int tt = t + k - 15;
                if (tt >= 0 && tt < Tn)
                    s += w0[k] * aw[tt] + w1[k] * awc[tt];
            }
            locS[tl * 32 + lane] = s;
        }
    }
    __syncthreads();

    // ---- location FC via WMMA: [16 x 32] x [32 x 128] ----
    const int l15   = lane & 15;
    const int khalf = lane >> 4;
    v8f acc[8];
#pragma unroll
    for (int j = 0; j < 8; ++j) acc[j] = (v8f){};

    for (int k = 0; k < 32; k += 4) {
        const int kk = k + 2 * khalf;
        v2f a = *(const v2f*)(locS + l15 * 32 + kk);
#pragma unroll
        for (int j = 0; j < 8; ++j) {
            v2f bb = *(const v2f*)(W_lfc + (size_t)(j * 16 + l15) * 32 + kk);
            acc[j] = __builtin_amdgcn_wmma_f32_16x16x4_f32(false, a, false, bb, (short)0, acc[j], false, false);
        }
    }

    // ---- energy: tanh + Wv dot, reduce across the a dimension ----
    float rowsum[8];
#pragma unroll
    for (int v = 0; v < 8; ++v) rowsum[v] = 0.0f;

#pragma unroll
    for (int j = 0; j < 8; ++j) {
        const int a_idx = j * 16 + l15;
        const float pqv = pq[(size_t)b * An + a_idx];
        const float wv  = Wv[a_idx];
#pragma unroll
        for (int v = 0; v < 8; ++v) {
            const int t = t0 + v + 8 * khalf;
            float val = acc[j][v] + pqv + pm[((size_t)b * Tn + t) * An + a_idx];
            float c = wv * tanhf(val);
            for (int off = 1; off < 16; off <<= 1)
                c += __shfl_xor(c, off, 32);
            rowsum[v] += c;
        }
    }

    if (l15 == 0) {
#pragma unroll
        for (int v = 0; v < 8; ++v) {
            const int t = t0 + v + 8 * khalf;
            float e = rowsum[v];
            if (mask[(size_t)b * Tn + t]) e = -1e30f;
            energies[(size_t)b * Tn + t] = e;
        }
    }
}

// ---------------------------------------------------------------------------
// Row softmax over T=2048
// ---------------------------------------------------------------------------
__global__ void __launch_bounds__(256)
softmax_rows(const float* __restrict__ energies, float* __restrict__ weights)
{
    __shared__ float red[256];
    const int b = blockIdx.x, tid = threadIdx.x;
    const float* e = energies + (size_t)b * Tn;

    float m = -3.4e38f;
    for (int t = tid; t < Tn; t += 256) m = fmaxf(m, e[t]);
    red[tid] = m; __syncthreads();
    for (int s = 128; s > 0; s >>= 1) {
        if (tid < s) red[tid] = fmaxf(red[tid], red[tid + s]);
        __syncthreads();
    }
    m = red[0]; __syncthreads();

    float sum = 0.0f;
    for (int t = tid; t < Tn; t += 256) sum += __expf(e[t] - m);
    red[tid] = sum; __syncthreads();
    for (int s = 128; s > 0; s >>= 1) {
        if (tid < s) red[tid] += red[tid + s];
        __syncthreads();
    }
    const float inv = 1.0f / red[0];

    for (int t = tid; t < Tn; t += 256)
        weights[(size_t)b * Tn + t] = __expf(e[t] - m) * inv;
}

// ---------------------------------------------------------------------------
// Partial context: ctx_part[chunk][b, e] = sum_{t in chunk} w[b,t]*memory[b,t,e]
// ---------------------------------------------------------------------------
__global__ void __launch_bounds__(256)
context_part(const float* __restrict__ weights, const float* __restrict__ memory,
             float* __restrict__ ctx_part)
{
    __shared__ float w[Tn / CTX_SPLIT];
    const int b = blockIdx.x, chunk = blockIdx.y, tid = threadIdx.x;
    const int tb = chunk * (Tn / CTX_SPLIT);
    for (int t = tid; t < Tn / CTX_SPLIT; t += 256)
        w[t] = weights[(size_t)b * Tn + tb + t];
    __syncthreads();
    float acc = 0.0f;
    const float* mb = memory + ((size_t)b * Tn + tb) * En + tid;
    for (int t = 0; t < Tn / CTX_SPLIT; ++t) acc += w[t] * mb[(size_t)t * En];
    ctx_part[((size_t)chunk * Bn + b) * En + tid] = acc;
}

// ---------------------------------------------------------------------------
// gate: stop[b] = dot(dhac[b,:1280], W_gate) + b_gate
// ---------------------------------------------------------------------------
__global__ void __launch_bounds__(32)
gate_kernel(const float* __restrict__ dhac, const float* __restrict__ Wg,
            const float* __restrict__ bg, float* __restrict__ out)
{
    const int b = blockIdx.x, lane = threadIdx.x;
    const float* x = dhac + (size_t)b * (Dn + En);
    float s = 0.0f;
    for (int k = lane; k < Dn + En; k += 32) s += x[k] * Wg[k];
    for (int off = 16; off > 0; off >>= 1) s += __shfl_xor(s, off, 32);
    if (lane == 0) out[b] = s + bg[0];
}

// ---------------------------------------------------------------------------
extern "C" void kernel_launch(void* const* d_in, const int* in_sizes, int n_in,
                              void* d_out, int out_size, void* d_ws, size_t ws_size,
                              hipStream_t stream)
{
    const float* last_frame = (const float*)d_in[0];
    const float* att_h      = (const float*)d_in[1];
    const float* att_c      = (const float*)d_in[2];
    const float* att_w      = (const float*)d_in[3];
    const float* att_w_cum  = (const float*)d_in[4];
    const float* att_ctx    = (const float*)d_in[5];
    const float* dec_h      = (const float*)d_in[6];
    const float* dec_c      = (const float*)d_in[7];
    const float* memory     = (const float*)d_in[8];
    const float* proc_mem   = (const float*)d_in[9];
    const float* W_ih_a     = (const float*)d_in[10];
    const float* W_hh_a     = (const float*)d_in[11];
    const float* b_ih_a     = (const float*)d_in[12];
    const float* b_hh_a     = (const float*)d_in[13];
    const float* W_q        = (const float*)d_in[14];
    const float* W_v        = (const float*)d_in[15];
    const float* W_loc      = (const float*)d_in[16];
    const float* W_lfc      = (const float*)d_in[17];
    const float* W_ih_d     = (const float*)d_in[18];
    const float* W_hh_d     = (const float*)d_in[19];
    const float* b_ih_d     = (const float*)d_in[20];
    const float* b_hh_d     = (const float*)d_in[21];
    const float* W_proj     = (const float*)d_in[22];
    const float* b_proj     = (const float*)d_in[23];
    const float* W_gate     = (const float*)d_in[24];
    const float* b_gate     = (const float*)d_in[25];
    const int*   mask       = (const int*)d_in[26];

    float* out      = (float*)d_out;
    float* out_stop = out + Bn * 256;              // 16384
    float* out_wts  = out_stop + Bn;               // 16448

    // workspace layout (floats)
    float* ws = (float*)d_ws;
    float* xcat_a    = ws;                                     // 64*1408
    float* gates_a   = xcat_a    + Bn * 1408;                  // 4 * 64*4096
    float* ah        = gates_a   + GEMM_SPLIT * Bn * 4096;     // 64*1024
    float* pq_part   = ah        + Bn * 1024;                  // 4 * 64*128
    float* pq        = pq_part   + GEMM_SPLIT * Bn * An;       // 64*128
    float* energ     = pq        + Bn * An;                    // 64*2048
    float* ctx_part  = energ     + Bn * Tn;                    // 8 * 64*256
    float* ctx       = ctx_part  + CTX_SPLIT * Bn * En;        // 64*256
    float* xcat_d    = ctx       + Bn * En;                    // 64*2304
    float* gates_d   = xcat_d    + Bn * 2304;                  // 4 * 64*4096
    float* dh        = gates_d   + GEMM_SPLIT * Bn * 4096;     // 64*1024
    float* dhac      = dh        + Bn * 1024;                  // 64*1280
    float* proj_part = dhac      + Bn * 1280;                  // 4 * 64*256

    // 1) attention LSTM
    {
        int tot = Bn * 1408;
        pack3<<<(tot + 255) / 256, 256, 0, stream>>>(xcat_a, last_frame, 128, att_ctx, 256, att_h, 1024);
        gemm_wmma_f32<1408, 384, 4096, GEMM_SPLIT, true>
            <<<dim3(4096 / 16, GEMM_SPLIT), 32, 0, stream>>>(
                xcat_a, W_ih_a, W_hh_a, b_ih_a, b_hh_a, gates_a);
        lstm_pointwise<<<(Bn * Rn + 255) / 256, 256, 0, stream>>>(gates_a, att_c, ah, Rn, GEMM_SPLIT);
    }

    // 2) processed query  pq = ah @ W_q^T
    gemm_wmma_f32<1024, 1024, 128, GEMM_SPLIT, false>
        <<<dim3(An / 16, GEMM_SPLIT), 32, 0, stream>>>(
            ah, W_q, nullptr, nullptr, nullptr, pq_part);
    reduce_partials<<<(Bn * An + 255) / 256, 256, 0, stream>>>(pq, pq_part, GEMM_SPLIT, Bn * An);

    // 3) fused location conv + FC + energies
    attn_energy<<<dim3(Tn / 16, Bn), 32, 0, stream>>>(att_w, att_w_cum, W_loc, W_lfc,
                                                      pq, proc_mem, W_v, mask, energ);

    // 4) softmax -> weights (directly into d_out)
    softmax_rows<<<Bn, 256, 0, stream>>>(energ, out_wts);

    // 5) context (T-split partials, then deterministic reduce)
    context_part<<<dim3(Bn, CTX_SPLIT), 256, 0, stream>>>(out_wts, memory, ctx_part);
    reduce_partials<<<(Bn * En + 255) / 256, 256, 0, stream>>>(ctx, ctx_part, CTX_SPLIT, Bn * En);

    // 6) decoder LSTM
    {
        int tot = Bn * 2304;
        pack3<<<(tot + 255) / 256, 256, 0, stream>>>(xcat_d, ah, 1024, ctx, 256, dec_h, 1024);
        gemm_wmma_f32<2304, 1280, 4096, GEMM_SPLIT, true>
            <<<dim3(4096 / 16, GEMM_SPLIT), 32, 0, stream>>>(
                xcat_d, W_ih_d, W_hh_d, b_ih_d, b_hh_d, gates_d);
        lstm_pointwise<<<(Bn * Dn + 255) / 256, 256, 0, stream>>>(gates_d, dec_c, dh, Dn, GEMM_SPLIT);
    }

    // 7) heads
    {
        int tot = Bn * 1280;
        pack3<<<(tot + 255) / 256, 256, 0, stream>>>(dhac, dh, 1024, ctx, 256, nullptr, 0);
        gemm_wmma_f32<1280, 1280, 256, GEMM_SPLIT, false>
            <<<dim3(256 / 16, GEMM_SPLIT), 32, 0, stream>>>(
                dhac, W_proj, nullptr, b_proj, nullptr, proj_part);
        reduce_partials<<<(Bn * 256 + 255) / 256, 256, 0, stream>>>(out, proj_part, GEMM_SPLIT, Bn * 256);
        gate_kernel<<<Bn, 32, 0, stream>>>(dhac, W_gate, b_gate, out_stop);
    }
}